// get_model_15401752723788
// MI455X (gfx1250) — compile-verified
//
#include <hip/hip_runtime.h>
#include <limits.h>

// Problem constants from the reference
#define NB     64          // batches
#define NCH    6           // channels in x (we use 0 and 1)
#define NPTS   131072      // points per batch
#define SDIM   128         // image side
#define SCALEF 62.0f       // S//2 - 2

// Work partitioning: one block covers one (batch, chunk) slice of points.
#define CHUNKS        32
#define PTS_PER_CHUNK (NPTS / CHUNKS)      // 4096 points
#define V4_PER_CHUNK  (PTS_PER_CHUNK / 4)  // 1024 float4 per channel
#define THREADS       256

// ---------------------------------------------------------------------------
// Kernel 1: init per-(batch,channel) mins to INT_MAX (128 ints in d_ws)
// ---------------------------------------------------------------------------
__global__ void init_mins_kernel(int* __restrict__ mins) {
  int i = threadIdx.x;
  if (i < 2 * NB) mins[i] = INT_MAX;
}

// ---------------------------------------------------------------------------
// Kernel 2: zero the output image (1 Mi floats -> 256 Ki float4)
// ---------------------------------------------------------------------------
__global__ void __launch_bounds__(THREADS) zero_out_kernel(float4* __restrict__ out) {
  size_t i = (size_t)blockIdx.x * blockDim.x + threadIdx.x;
  out[i] = make_float4(0.f, 0.f, 0.f, 0.f);
}

// ---------------------------------------------------------------------------
// Kernel 3: per-(batch,channel) min of trunc(x*62) over all points.
// float4 (global_load_b128) streaming, wave32 butterfly reduce, one
// global_atomic_min_i32 per wave per channel.
// ---------------------------------------------------------------------------
__global__ void __launch_bounds__(THREADS) min_kernel(const float* __restrict__ x,
                                                      int* __restrict__ mins) {
  const int b     = blockIdx.x / CHUNKS;
  const int chunk = blockIdx.x % CHUNKS;

  const float4* __restrict__ x0 =
      (const float4*)(x + ((size_t)b * NCH + 0) * NPTS) + (size_t)chunk * V4_PER_CHUNK;
  const float4* __restrict__ x1 =
      (const float4*)(x + ((size_t)b * NCH + 1) * NPTS) + (size_t)chunk * V4_PER_CHUNK;

  int m0 = INT_MAX, m1 = INT_MAX;

#pragma unroll
  for (int k = 0; k < V4_PER_CHUNK; k += THREADS) {
    const int idx = k + threadIdx.x;
    // gfx1250 prefetch path (global_prefetch_b8); speculative, OOB is dropped.
    __builtin_prefetch(&x0[idx + 2 * THREADS], 0, 0);
    __builtin_prefetch(&x1[idx + 2 * THREADS], 0, 0);
    float4 a = x0[idx];
    float4 c = x1[idx];
    // (int) cast = v_cvt_i32_f32 round-toward-zero == jnp.trunc + astype(int32)
    m0 = min(m0, (int)(a.x * SCALEF));
    m0 = min(m0, (int)(a.y * SCALEF));
    m0 = min(m0, (int)(a.z * SCALEF));
    m0 = min(m0, (int)(a.w * SCALEF));
    m1 = min(m1, (int)(c.x * SCALEF));
    m1 = min(m1, (int)(c.y * SCALEF));
    m1 = min(m1, (int)(c.z * SCALEF));
    m1 = min(m1, (int)(c.w * SCALEF));
  }

  // wave32 butterfly reduction (lane-permute hardware, no LDS memory)
#pragma unroll
  for (int off = 16; off > 0; off >>= 1) {
    m0 = min(m0, __shfl_xor(m0, off, 32));
    m1 = min(m1, __shfl_xor(m1, off, 32));
  }

  if ((threadIdx.x & 31) == 0) {
    atomicMin(&mins[b * 2 + 0], m0);  // global_atomic_min_i32
    atomicMin(&mins[b * 2 + 1], m1);
  }
}

// ---------------------------------------------------------------------------
// Kernel 4: scatter-add pass. Re-reads the (now L2-resident) 64MB of channel
// data, recomputes cl / weight, bound-checks shifted coords, and does a
// native fp32 global atomic add into the per-batch 128x128 image.
// ---------------------------------------------------------------------------
__device__ __forceinline__ void splat_one(float p0raw, float p1raw,
                                          int mn0, int mn1,
                                          float* __restrict__ img) {
  const float p0 = p0raw * SCALEF;
  const float p1 = p1raw * SCALEF;
  const int cl0 = (int)p0;               // trunc toward zero
  const int cl1 = (int)p1;
  const float f0 = fabsf((float)cl0 - p0);
  const float f1 = fabsf((float)cl1 - p1);
  const int c0 = cl0 - mn0;              // >= 0 by construction (mn is the min)
  const int c1 = cl1 - mn1;
  if ((unsigned)c0 < SDIM && (unsigned)c1 < SDIM) {
    const float w = 2.0f - (f0 + f1);    // match reference: sum abs, then 2 - sum
#ifdef __HIP_PLATFORM_AMD__
    unsafeAtomicAdd(&img[c0 * SDIM + c1], w);   // global_atomic_add_f32, no return
#else
    atomicAdd(&img[c0 * SDIM + c1], w);
#endif
  }
}

__global__ void __launch_bounds__(THREADS) splat_kernel(const float* __restrict__ x,
                                                        const int* __restrict__ mins,
                                                        float* __restrict__ out) {
  const int b     = blockIdx.x / CHUNKS;
  const int chunk = blockIdx.x % CHUNKS;

  const int mn0 = mins[b * 2 + 0];   // uniform -> scalar loads
  const int mn1 = mins[b * 2 + 1];

  float* __restrict__ img = out + (size_t)b * SDIM * SDIM;

  const float4* __restrict__ x0 =
      (const float4*)(x + ((size_t)b * NCH + 0) * NPTS) + (size_t)chunk * V4_PER_CHUNK;
  const float4* __restrict__ x1 =
      (const float4*)(x + ((size_t)b * NCH + 1) * NPTS) + (size_t)chunk * V4_PER_CHUNK;

#pragma unroll
  for (int k = 0; k < V4_PER_CHUNK; k += THREADS) {
    const int idx = k + threadIdx.x;
    __builtin_prefetch(&x0[idx + 2 * THREADS], 0, 0);
    __builtin_prefetch(&x1[idx + 2 * THREADS], 0, 0);
    float4 a = x0[idx];
    float4 c = x1[idx];
    splat_one(a.x, c.x, mn0, mn1, img);
    splat_one(a.y, c.y, mn0, mn1, img);
    splat_one(a.z, c.z, mn0, mn1, img);
    splat_one(a.w, c.w, mn0, mn1, img);
  }
}

// ---------------------------------------------------------------------------
// Launcher
// ---------------------------------------------------------------------------
extern "C" void kernel_launch(void* const* d_in, const int* in_sizes, int n_in,
                              void* d_out, int out_size, void* d_ws, size_t ws_size,
                              hipStream_t stream) {
  const float* x = (const float*)d_in[0];   // (64, 6, 131072) f32
  float* out = (float*)d_out;               // (64, 1, 128, 128) f32 flat
  int* mins = (int*)d_ws;                   // 128 ints of scratch

  init_mins_kernel<<<1, 128, 0, stream>>>(mins);
  zero_out_kernel<<<(NB * SDIM * SDIM / 4) / THREADS, THREADS, 0, stream>>>((float4*)out);
  min_kernel<<<NB * CHUNKS, THREADS, 0, stream>>>(x, mins);
  splat_kernel<<<NB * CHUNKS, THREADS, 0, stream>>>(x, mins, out);
}